// CTCCenterLoss_19035295056206
// MI455X (gfx1250) — compile-verified
//
#include <hip/hip_runtime.h>

#define ALPHA 0.5f
#define C 86
#define F 256
#define NROWS 131072
#define BLOCKS 256
#define ROWS_PER_BLOCK (NROWS / BLOCKS)          /* 512 */
#define TILE_ROWS 32
#define TILES_PER_BLOCK (ROWS_PER_BLOCK / TILE_ROWS) /* 16 */
#define THREADS 256

typedef __attribute__((ext_vector_type(4))) unsigned int u32x4;
typedef __attribute__((ext_vector_type(8))) int i32x8;
typedef __attribute__((ext_vector_type(4))) int i32x4;

// Build a Tensor DMA Descriptor (D#) for a 2D f32 tile [TILE_ROWS x F]
// (row stride F elements) from global memory into LDS, and issue the
// TDM load. 6-arg builtin form (clang-23 / therock-10.0 headers).
// Tracked with TENSORcnt.
__device__ __forceinline__ void tdm_load_tile(const float* gsrc,
                                              unsigned lds_byte_addr) {
    unsigned long long ga = (unsigned long long)(size_t)gsrc;
    u32x4 g0;
    g0[0] = 1u;                                   // count=1 (valid), user mode
    g0[1] = lds_byte_addr;                        // lds_addr [63:32]
    g0[2] = (unsigned)(ga & 0xFFFFFFFFull);       // global_addr [95:64]
    g0[3] = (unsigned)((ga >> 32) & 0x1FFFFFFull) // global_addr [120:96]
          | (2u << 30);                           // type=2 ("image")
    i32x8 g1;
    g1[0] = (int)(2u << 16);                      // data_size=2 -> 4 bytes
    g1[1] = (int)(((unsigned)F & 0xFFFFu) << 16); // tensor_dim0[15:0] @ 63:48
    g1[2] = (int)(((unsigned)F >> 16)             // tensor_dim0[31:16]
          | (((unsigned)TILE_ROWS & 0xFFFFu) << 16)); // tensor_dim1[15:0]
    g1[3] = (int)(((unsigned)TILE_ROWS >> 16)     // tensor_dim1[31:16]
          | ((unsigned)F << 16));                 // tile_dim0 = 256
    g1[4] = (int)TILE_ROWS;                       // tile_dim1 = 32, tile_dim2=0
    g1[5] = (int)F;                               // tensor_dim0_stride lo32
    g1[6] = 0;                                    // stride0 hi16, stride1 lo16
    g1[7] = 0;
    i32x4 gz4;
    gz4[0] = 0; gz4[1] = 0; gz4[2] = 0; gz4[3] = 0; // groups 2/3 unused (2D)
    i32x8 gz8;
    gz8[0] = 0; gz8[1] = 0; gz8[2] = 0; gz8[3] = 0;
    gz8[4] = 0; gz8[5] = 0; gz8[6] = 0; gz8[7] = 0; // extra group (unused)
    __builtin_amdgcn_tensor_load_to_lds(g0, g1, gz4, gz4, gz8, 0);
}

__global__ void init_ws(float* __restrict__ ws, float* __restrict__ out) {
    int i = blockIdx.x * blockDim.x + threadIdx.x;
    if (i < C * F + C + 1) ws[i] = 0.0f;
    if (i == 0) out[0] = 0.0f;
}

__global__ __launch_bounds__(THREADS)
void center_accum(const float* __restrict__ feat, const int* __restrict__ labels,
                  float* __restrict__ ws) {
    extern __shared__ float smem[];
    float* acc  = smem;                       // C*F floats (88064 B)
    float* cnt  = acc + C * F;                // 128 floats (86 used)
    int*   lab  = (int*)(cnt + 128);          // ROWS_PER_BLOCK ints
    float* tiles = (float*)(lab + ROWS_PER_BLOCK); // 2 x TILE_ROWS*F floats

    const int tid  = threadIdx.x;
    const int wave = tid >> 5;
    const int lane = tid & 31;
    const int rowBase = blockIdx.x * ROWS_PER_BLOCK;

    for (int i = tid; i < C * F; i += THREADS) acc[i] = 0.0f;
    if (tid < C) cnt[tid] = 0.0f;
    for (int i = tid; i < ROWS_PER_BLOCK; i += THREADS)
        lab[i] = labels[rowBase + i];

    // Prologue: wave 0 kicks off TDM DMA of tile 0 into buffer 0.
    if (tid < 32) {
        tdm_load_tile(feat + (size_t)rowBase * F,
                      (unsigned)(size_t)(void*)tiles);
    }
    __syncthreads();

    float sumsq = 0.0f;

    for (int t = 0; t < TILES_PER_BLOCK; ++t) {
        if (tid < 32) {
            if (t + 1 < TILES_PER_BLOCK) {
                // double buffer: issue next tile, then wait for current
                tdm_load_tile(feat + (size_t)(rowBase + (t + 1) * TILE_ROWS) * F,
                              (unsigned)(size_t)(void*)(tiles + ((t + 1) & 1) * TILE_ROWS * F));
                __builtin_amdgcn_s_wait_tensorcnt(1);
            } else {
                __builtin_amdgcn_s_wait_tensorcnt(0);
            }
        }
        __syncthreads();   // tile t visible to whole workgroup

        const float* tb = tiles + (t & 1) * TILE_ROWS * F;
        for (int r = wave; r < TILE_ROWS; r += THREADS / 32) {
            const int label = lab[t * TILE_ROWS + r];
            const float* rowp = tb + r * F;
            float* accp = acc + label * F;
            #pragma unroll
            for (int j = 0; j < F / 32; ++j) {
                // lane l owns columns l, l+32, ... -> bank-conflict-free
                const int col = lane + 32 * j;
                float v = rowp[col];
                sumsq += v * v;
                __hip_atomic_fetch_add(&accp[col], v, __ATOMIC_RELAXED,
                                       __HIP_MEMORY_SCOPE_WORKGROUP);
            }
            if (lane == 0)
                __hip_atomic_fetch_add(&cnt[label], 1.0f, __ATOMIC_RELAXED,
                                       __HIP_MEMORY_SCOPE_WORKGROUP);
        }
        __syncthreads();   // done with buffer before TDM reuses it
    }

    // Flush block-local accumulator to global workspace.
    float* gacc = ws;
    float* gcnt = ws + C * F;
    float* gsq  = ws + C * F + C;
    for (int i = tid; i < C * F; i += THREADS)
        unsafeAtomicAdd(&gacc[i], acc[i]);
    if (tid < C) unsafeAtomicAdd(&gcnt[tid], cnt[tid]);

    float* red = tiles;    // reuse tile buffer as reduction scratch
    red[tid] = sumsq;
    __syncthreads();
    for (int s = THREADS / 2; s > 0; s >>= 1) {
        if (tid < s) red[tid] += red[tid + s];
        __syncthreads();
    }
    if (tid == 0) unsafeAtomicAdd(gsq, red[0]);
}

__global__ __launch_bounds__(F)
void finalize(const float* __restrict__ ws, const float* __restrict__ centers,
              float* __restrict__ out) {
    __shared__ float red[F];
    const int c = blockIdx.x;     // class
    const int f = threadIdx.x;    // feature column
    float S   = ws[c * F + f];
    float n   = ws[C * F + c];
    float cen = centers[c * F + f];
    out[1 + c * F + f] = cen + ALPHA * (S - n * cen) / (1.0f + n);
    // loss contribution: 0.5*n*cen^2 - S*cen  (plus 0.5*sumsq once)
    red[f] = 0.5f * n * cen * cen - S * cen;
    __syncthreads();
    for (int s = F / 2; s > 0; s >>= 1) {
        if (f < s) red[f] += red[f + s];
        __syncthreads();
    }
    if (f == 0) {
        float v = red[0];
        if (c == 0) v += 0.5f * ws[C * F + C];
        unsafeAtomicAdd(&out[0], v);
    }
}

extern "C" void kernel_launch(void* const* d_in, const int* in_sizes, int n_in,
                              void* d_out, int out_size, void* d_ws, size_t ws_size,
                              hipStream_t stream) {
    (void)in_sizes; (void)n_in; (void)out_size; (void)ws_size;
    const float* feat    = (const float*)d_in[0];
    const int*   labels  = (const int*)d_in[1];
    const float* centers = (const float*)d_in[2];
    float* out = (float*)d_out;
    float* ws  = (float*)d_ws;

    init_ws<<<(C * F + C + 1 + 255) / 256, 256, 0, stream>>>(ws, out);

    size_t lds_bytes = (size_t)(C * F + 128) * 4   // acc + cnt
                     + (size_t)ROWS_PER_BLOCK * 4  // labels
                     + (size_t)2 * TILE_ROWS * F * 4; // double-buffered tiles
    center_accum<<<BLOCKS, THREADS, lds_bytes, stream>>>(feat, labels, ws);

    finalize<<<C, F, 0, stream>>>(ws, centers, out);
}